// HeteroGNN_29746943492593
// MI455X (gfx1250) — compile-verified
//
#include <hip/hip_runtime.h>

typedef __attribute__((ext_vector_type(2))) float v2f;
typedef __attribute__((ext_vector_type(8))) float v8f;

#define NA 50000
#define NB 50000
#define NE 1600000
#define D 128      // D_IN == HID
#define OUTC 64
#define STRIPS (NA / 16)   // 3125, exact (50000 = 16*3125)

// ---------------- utility kernels ----------------

__global__ void zero_f32(float* __restrict__ p, int n4) {
    int i = blockIdx.x * blockDim.x + threadIdx.x;
    if (i < n4) {
        float4 z = {0.f, 0.f, 0.f, 0.f};
        *(float4*)(p + (size_t)i * 4) = z;
    }
}

__global__ void count_deg(const int* __restrict__ dst, float* __restrict__ cnt, int e) {
    int i = blockIdx.x * blockDim.x + threadIdx.x;
    if (i < e) atomicAdd(&cnt[dst[i]], 1.0f);
}

// one wave per edge: 32 lanes x float4 = 128 features
__global__ void scatter_sum(const int* __restrict__ src, const int* __restrict__ dst,
                            const float* __restrict__ h, float* __restrict__ sum, int e) {
    int t    = blockIdx.x * blockDim.x + threadIdx.x;
    int edge = t >> 5;
    int lane = t & 31;
    if (edge >= e) return;
    int s = src[edge];
    int d = dst[edge];
    const float4 v = *(const float4*)(h + (size_t)s * D + lane * 4);
    float* o = sum + (size_t)d * D + lane * 4;
    atomicAdd(o + 0, v.x);
    atomicAdd(o + 1, v.y);
    atomicAdd(o + 2, v.z);
    atomicAdd(o + 3, v.w);
}

// ---------------- fused dual-GEMM with fp32 WMMA ----------------
// out[row, :] = act( scale(X1[row,:]) @ W1 [+ X2[row,:] @ W2] + bias )
// X* : [nrows x 128] row-major, W* : [128 x (TN*16)] row-major.
// One wave owns a 16-row strip and all TN column tiles (A reused across tiles).
// DUAL/SCALED are compile-time so the WMMA region is branch-free straight-line
// code: EXEC is never touched around v_wmma (ISA requires EXEC == all-1s).
template <int TN, bool DUAL, bool SCALED>
__global__ void gemm_strip(const float* __restrict__ X1, const float* __restrict__ cnt,
                           const float* __restrict__ W1,
                           const float* __restrict__ X2, const float* __restrict__ W2,
                           const float* __restrict__ bias, float* __restrict__ out,
                           int nstrips, int do_relu) {
    const int wave = (blockIdx.x * blockDim.x + threadIdx.x) >> 5;
    if (wave >= nstrips) return;                 // wave-uniform: EXEC stays all-1s
    const int lane = threadIdx.x & 31;
    const int half = lane >> 4;                  // selects K pair {0,1} vs {2,3}
    const int l16  = lane & 15;
    const int row  = wave * 16 + l16;            // A-frag: M = lane&15 for both halves
    const int NC   = TN * 16;

    float inv = 1.0f;
    if (SCALED) inv = 1.0f / fmaxf(cnt[row], 1.0f); // fold mean-division into A load

    v8f acc[TN] = {};

    const float* x1r = X1 + (size_t)row * D;
    const float* x2r = DUAL ? (X2 + (size_t)row * D) : nullptr;

    for (int k0 = 0; k0 < D; k0 += 4) {
        const int ka = k0 + 2 * half;            // even -> aligned float2
        v2f a1 = *(const v2f*)(x1r + ka);
        if (SCALED) a1 *= inv;
#pragma unroll
        for (int j = 0; j < TN; ++j) {
            const int col = j * 16 + l16;        // B-frag: N = lane&15, K = ka+{0,1}
            v2f b;
            b.x = W1[ka * NC + col];
            b.y = W1[(ka + 1) * NC + col];
            acc[j] = __builtin_amdgcn_wmma_f32_16x16x4_f32(
                false, a1, false, b, (short)0, acc[j], false, false);
        }
        if (DUAL) {
            v2f a2 = *(const v2f*)(x2r + ka);
#pragma unroll
            for (int j = 0; j < TN; ++j) {
                const int col = j * 16 + l16;
                v2f b;
                b.x = W2[ka * NC + col];
                b.y = W2[(ka + 1) * NC + col];
                acc[j] = __builtin_amdgcn_wmma_f32_16x16x4_f32(
                    false, a2, false, b, (short)0, acc[j], false, false);
            }
        }
    }

#pragma unroll
    for (int j = 0; j < TN; ++j) {
        const int col = j * 16 + l16;
        const float bv = bias[col];
#pragma unroll
        for (int i = 0; i < 8; ++i) {            // C/D: VGPR i -> row i + 8*half
            float v = acc[j][i] + bv;
            if (do_relu) v = fmaxf(v, 0.0f);
            out[(size_t)(wave * 16 + i + 8 * half) * NC + col] = v;
        }
    }
}

// ---------------- launch ----------------

extern "C" void kernel_launch(void* const* d_in, const int* in_sizes, int n_in,
                              void* d_out, int out_size, void* d_ws, size_t ws_size,
                              hipStream_t stream) {
    const float* x_a   = (const float*)d_in[0];
    const float* x_b   = (const float*)d_in[1];
    const int*   ei_ab = (const int*)d_in[2];    // [2, E]: row0 src(a), row1 dst(b)
    const int*   ei_ba = (const int*)d_in[3];
    const float* Wa_in = (const float*)d_in[4];
    const float* ba_in = (const float*)d_in[5];
    const float* Wb_in = (const float*)d_in[6];
    const float* bb_in = (const float*)d_in[7];
    const float* Wl_ab = (const float*)d_in[8];  // [2,128,128]
    const float* Wr_ab = (const float*)d_in[9];
    const float* bl_ab = (const float*)d_in[10]; // [2,128]
    const float* Wl_ba = (const float*)d_in[11];
    const float* Wr_ba = (const float*)d_in[12];
    const float* bl_ba = (const float*)d_in[13];
    const float* W_out = (const float*)d_in[14]; // [128,64]
    const float* b_out = (const float*)d_in[15];
    float*       out   = (float*)d_out;          // [50000,64]

    const size_t FEAT = (size_t)NA * D;          // 6.4M floats per buffer
    float* ws   = (float*)d_ws;
    float* hA0  = ws;
    float* hA1  = hA0 + FEAT;
    float* sumA = hA1 + FEAT;
    float* hB0  = sumA + FEAT;
    float* hB1  = hB0 + FEAT;
    float* sumB = hB1 + FEAT;
    float* cntA = sumB + FEAT;
    float* cntB = cntA + NA;

    const int TB = 256;
    const dim3 blk(TB);
    const int featB  = (int)((FEAT / 4 + TB - 1) / TB);            // zero feature buf
    const int cntB4  = (NA / 4 + TB - 1) / TB;                     // zero count buf
    const int edgeB  = (NE + TB - 1) / TB;                         // 1 thread / edge
    const int scatB  = (int)(((size_t)NE * 32 + TB - 1) / TB);     // 1 wave / edge
    const int gemmB  = (STRIPS * 32 + TB - 1) / TB;                // 1 wave / strip

    // degree counts (edge structure is constant across layers)
    zero_f32<<<cntB4, blk, 0, stream>>>(cntA, NA / 4);
    zero_f32<<<cntB4, blk, 0, stream>>>(cntB, NA / 4);
    count_deg<<<edgeB, blk, 0, stream>>>(ei_ab + NE, cntB, NE);    // dst in b
    count_deg<<<edgeB, blk, 0, stream>>>(ei_ba + NE, cntA, NE);    // dst in a

    // input projections: h = relu(x @ W_in + b_in)
    gemm_strip<8, false, false><<<gemmB, blk, 0, stream>>>(
        x_a, nullptr, Wa_in, nullptr, nullptr, ba_in, hA0, STRIPS, 1);
    gemm_strip<8, false, false><<<gemmB, blk, 0, stream>>>(
        x_b, nullptr, Wb_in, nullptr, nullptr, bb_in, hB0, STRIPS, 1);

    float *curA = hA0, *newA = hA1, *curB = hB0, *newB = hB1;
    for (int l = 0; l < 2; ++l) {
        zero_f32<<<featB, blk, 0, stream>>>(sumA, (int)(FEAT / 4));
        zero_f32<<<featB, blk, 0, stream>>>(sumB, (int)(FEAT / 4));
        scatter_sum<<<scatB, blk, 0, stream>>>(ei_ab, ei_ab + NE, curA, sumB, NE);
        scatter_sum<<<scatB, blk, 0, stream>>>(ei_ba, ei_ba + NE, curB, sumA, NE);
        // new_b = relu(mean_b @ Wl_ab + h_b @ Wr_ab + bl_ab)
        gemm_strip<8, true, true><<<gemmB, blk, 0, stream>>>(
            sumB, cntB, Wl_ab + l * D * D, curB, Wr_ab + l * D * D,
            bl_ab + l * D, newB, STRIPS, 1);
        // new_a = relu(mean_a @ Wl_ba + h_a @ Wr_ba + bl_ba)
        gemm_strip<8, true, true><<<gemmB, blk, 0, stream>>>(
            sumA, cntA, Wl_ba + l * D * D, curA, Wr_ba + l * D * D,
            bl_ba + l * D, newA, STRIPS, 1);
        float* t;
        t = curA; curA = newA; newA = t;
        t = curB; curB = newB; newB = t;
    }

    // final: out = h_a @ W_out + b_out  (no relu)
    gemm_strip<4, false, false><<<gemmB, blk, 0, stream>>>(
        curA, nullptr, W_out, nullptr, nullptr, b_out, out, STRIPS, 0);
}